// GCNNet_13073880449158
// MI455X (gfx1250) — compile-verified
//
#include <hip/hip_runtime.h>
#include <hip/hip_bf16.h>

#define DIM 133
#define KP  136          // K padded to multiple of 4
#define NP  144          // N (cols) padded to multiple of 16
#define BN_EPS 1e-5f

typedef float v2f __attribute__((ext_vector_type(2)));
typedef float v8f __attribute__((ext_vector_type(8)));

__device__ __forceinline__ void gAtomicAddF(float* p, float v) {
    // native global_atomic_add_f32 (device scope, relaxed) — avoid CAS loop
    __hip_atomic_fetch_add(p, v, __ATOMIC_RELAXED, __HIP_MEMORY_SCOPE_AGENT);
}

// ---------------------------------------------------------------- utilities
__global__ void k_fill(float* __restrict__ p, int n, float v) {
    int i = blockIdx.x * blockDim.x + threadIdx.x;
    if (i < n) p[i] = v;
}

// zero-padded copy of W (DIMxDIM) into Wp (KPxNP)
__global__ void k_padW(const float* __restrict__ W, float* __restrict__ Wp) {
    int i = blockIdx.x * blockDim.x + threadIdx.x;
    if (i >= KP * NP) return;
    int kk = i / NP;
    int cc = i - kk * NP;
    Wp[i] = (kk < DIM && cc < DIM) ? W[kk * DIM + cc] : 0.0f;
}

// deg[c] += 1 per edge (deg pre-initialized to 1.0 for the self-loop)
__global__ void k_deg(const int* __restrict__ col, int E, float* __restrict__ deg) {
    int e = blockIdx.x * blockDim.x + threadIdx.x;
    if (e < E) gAtomicAddF(&deg[col[e]], 1.0f);
}

__global__ void k_rsqrt_inplace(float* __restrict__ d, int n) {
    int i = blockIdx.x * blockDim.x + threadIdx.x;
    if (i < n) {
        float v = d[i];
        d[i] = (v > 0.0f) ? rsqrtf(v) : 0.0f;
    }
}

// ---------------------------------------------------------------- fp32 WMMA GEMM
// H[N,DIM] = X[N,DIM] @ W[DIM,DIM]  via V_WMMA_F32_16X16X4_F32
// Requires: nRows % 16 == 0 (holds: 50000 = 3125*16). Wp is the KPxNP padded W.
// One wave per 16x16 output tile; 8 waves per block.
__global__ void k_gemm_wmma(const float* __restrict__ X, const float* __restrict__ Wp,
                            float* __restrict__ H, int nRowTiles) {
    const int lane    = threadIdx.x & 31;
    const int wv      = threadIdx.x >> 5;
    const int rowTile = blockIdx.x * (blockDim.x >> 5) + wv;
    const int colTile = blockIdx.y;
    if (rowTile >= nRowTiles) return;          // wave-uniform exit; EXEC stays full

    const int half = lane >> 4;                // 0: K pair (k,k+1), 1: (k+2,k+3)
    const int sub  = lane & 15;
    const size_t rowBase = (size_t)(rowTile * 16 + sub) * DIM;  // A row base
    const int   colN  = colTile * 16 + sub;                     // B/D column
    const float* bcol = Wp + colN;             // always in-bounds (colN < NP)

    v8f acc = {};
    // main loop: ka up to 131 < DIM -> all loads unconditional
    for (int k = 0; k < 132; k += 4) {
        const int ka = k + half * 2;
        v2f a, b;
        a.x = X[rowBase + ka];
        a.y = X[rowBase + ka + 1];
        b.x = bcol[(size_t)ka * NP];
        b.y = bcol[(size_t)(ka + 1) * NP];
        acc = __builtin_amdgcn_wmma_f32_16x16x4_f32(
            false, a, false, b, (short)0, acc, false, false);
    }
    // tail k = 132: ka = 132 (half 0) or 134 (half 1); only ka==132 is a real
    // A element. Clamped address keeps the load in-bounds; value select zeros
    // the rest. B rows 132..135 exist in Wp (133..135 are zero padding).
    {
        const int ka = 132 + half * 2;
        const float ax = X[rowBase + (ka < DIM ? ka : DIM - 1)];
        v2f a, b;
        a.x = (ka < DIM) ? ax : 0.0f;
        a.y = 0.0f;                            // ka+1 >= 133 always
        b.x = bcol[(size_t)ka * NP];
        b.y = bcol[(size_t)(ka + 1) * NP];
        acc = __builtin_amdgcn_wmma_f32_16x16x4_f32(
            false, a, false, b, (short)0, acc, false, false);
    }

    if (colN < DIM) {                          // stores only; WMMAs already done
        #pragma unroll
        for (int j = 0; j < 8; ++j) {
            int r = rowTile * 16 + half * 8 + j;   // D layout: VGPR j -> M=j / M=8+j
            H[(size_t)r * DIM + colN] = acc[j];
        }
    }
}

// ---------------------------------------------------------------- edge scatter
// one wave32 per edge: OUT[c][:] += H[r][:] * dinv[r]*dinv[c]
__global__ void k_scatter(const float* __restrict__ H, const int* __restrict__ row,
                          const int* __restrict__ col, const float* __restrict__ dinv,
                          float* __restrict__ OUT, int E) {
    int wave = (int)((blockIdx.x * (size_t)blockDim.x + threadIdx.x) >> 5);
    int lane = threadIdx.x & 31;
    if (wave >= E) return;
    int r = row[wave];
    int c = col[wave];
    float nrm = dinv[r] * dinv[c];
    const float* hs = H + (size_t)r * DIM;
    float* od = OUT + (size_t)c * DIM;
    for (int d = lane; d < DIM; d += 32)
        gAtomicAddF(&od[d], hs[d] * nrm);
}

// ---------------------------------------------------------------- fused epilogue
// Acc[i] = BN(relu(Acc[i] + Hself[i]*dinv^2 + b))   (in place)
__global__ void k_fused(const float* __restrict__ Hself, float* __restrict__ Acc,
                        const float* __restrict__ dinv,
                        const float* __restrict__ b,  const float* __restrict__ g,
                        const float* __restrict__ beta, const float* __restrict__ rm,
                        const float* __restrict__ rv, int nNodes) {
    int i = blockIdx.x * blockDim.x + threadIdx.x;
    int tot = nNodes * DIM;
    if (i >= tot) return;
    int n = i / DIM;
    int d = i - n * DIM;
    float di = dinv[n];
    float v = Acc[i] + Hself[i] * di * di + b[d];
    v = fmaxf(v, 0.0f);
    v = (v - rm[d]) * rsqrtf(rv[d] + BN_EPS) * g[d] + beta[d];
    Acc[i] = v;
}

// ---------------------------------------------------------------- pooling
__global__ void k_pool_feat(const float* __restrict__ H, const int* __restrict__ batch,
                            float* __restrict__ sums, int nNodes) {
    int i = blockIdx.x * blockDim.x + threadIdx.x;
    int tot = nNodes * DIM;
    if (i >= tot) return;
    int n = i / DIM;
    int d = i - n * DIM;
    gAtomicAddF(&sums[(size_t)batch[n] * DIM + d], H[i]);
}

__global__ void k_pool_cnt(const int* __restrict__ batch, float* __restrict__ cnt, int nNodes) {
    int i = blockIdx.x * blockDim.x + threadIdx.x;
    if (i < nNodes) gAtomicAddF(&cnt[batch[i]], 1.0f);
}

__global__ void k_pool_fin(float* __restrict__ sums, const float* __restrict__ cnt, int nTot) {
    int i = blockIdx.x * blockDim.x + threadIdx.x;
    if (i < nTot) sums[i] = sums[i] / fmaxf(cnt[i / DIM], 1.0f);
}

// ---------------------------------------------------------------- launch
extern "C" void kernel_launch(void* const* d_in, const int* in_sizes, int n_in,
                              void* d_out, int out_size, void* d_ws, size_t ws_size,
                              hipStream_t stream) {
    const float* x    = (const float*)d_in[0];
    const int*   ei   = (const int*)  d_in[1];
    const int*   bat  = (const int*)  d_in[2];
    const float* W1   = (const float*)d_in[3];
    const float* b1   = (const float*)d_in[4];
    const float* W2   = (const float*)d_in[5];
    const float* b2   = (const float*)d_in[6];
    const float* g1   = (const float*)d_in[7];
    const float* be1  = (const float*)d_in[8];
    const float* rm1  = (const float*)d_in[9];
    const float* rv1  = (const float*)d_in[10];
    const float* g2   = (const float*)d_in[11];
    const float* be2  = (const float*)d_in[12];
    const float* rm2  = (const float*)d_in[13];
    const float* rv2  = (const float*)d_in[14];
    float*       out  = (float*)d_out;

    const int N  = in_sizes[0] / DIM;     // 50000
    const int E  = in_sizes[1] / 2;       // 1.6M
    const int nG = out_size / DIM;        // 2048
    const int* row = ei;                  // edge_index[0]
    const int* col = ei + E;              // edge_index[1]

    float*  ws   = (float*)d_ws;
    size_t  ND   = (size_t)N * DIM;
    float*  hA   = ws;                    // N*DIM
    float*  hB   = ws + ND;               // N*DIM
    float*  dinv = ws + 2 * ND;           // N
    float*  cnt  = dinv + N;              // nG
    float*  Wp1  = cnt + nG;              // KP*NP
    float*  Wp2  = Wp1 + (size_t)KP * NP; // KP*NP

    const int T = 256;
    auto cdiv = [](long a, long b) { return (int)((a + b - 1) / b); };

    const int nRowTiles = (N + 15) / 16;          // exact: 3125
    dim3 ggrid(cdiv(nRowTiles, 8), NP / 16);      // 9 column tiles

    // padded weights
    k_padW<<<cdiv(KP * NP, T), T, 0, stream>>>(W1, Wp1);
    k_padW<<<cdiv(KP * NP, T), T, 0, stream>>>(W2, Wp2);

    // degrees -> dinv
    k_fill<<<cdiv(N, T), T, 0, stream>>>(dinv, N, 1.0f);          // self-loop
    k_deg<<<cdiv(E, T), T, 0, stream>>>(col, E, dinv);
    k_rsqrt_inplace<<<cdiv(N, T), T, 0, stream>>>(dinv, N);

    // ---- layer 1 ----
    k_gemm_wmma<<<ggrid, T, 0, stream>>>(x, Wp1, hA, nRowTiles);  // hA = x@W1
    k_fill<<<cdiv((long)ND, T), T, 0, stream>>>(hB, (int)ND, 0.0f);
    k_scatter<<<cdiv(E, 8), T, 0, stream>>>(hA, row, col, dinv, hB, E);
    k_fused<<<cdiv((long)ND, T), T, 0, stream>>>(hA, hB, dinv, b1, g1, be1, rm1, rv1, N);

    // ---- layer 2 ----
    k_gemm_wmma<<<ggrid, T, 0, stream>>>(hB, Wp2, hA, nRowTiles); // hA = h1@W2
    k_fill<<<cdiv((long)ND, T), T, 0, stream>>>(hB, (int)ND, 0.0f);
    k_scatter<<<cdiv(E, 8), T, 0, stream>>>(hA, row, col, dinv, hB, E);
    k_fused<<<cdiv((long)ND, T), T, 0, stream>>>(hA, hB, dinv, b2, g2, be2, rm2, rv2, N);

    // ---- global mean pool ----
    k_fill<<<cdiv((long)nG * DIM, T), T, 0, stream>>>(out, nG * DIM, 0.0f);
    k_fill<<<cdiv(nG, T), T, 0, stream>>>(cnt, nG, 0.0f);
    k_pool_feat<<<cdiv((long)ND, T), T, 0, stream>>>(hB, bat, out, N);
    k_pool_cnt<<<cdiv(N, T), T, 0, stream>>>(bat, cnt, N);
    k_pool_fin<<<cdiv((long)nG * DIM, T), T, 0, stream>>>(out, cnt, nG * DIM);
}